// UpsampleF_89584427860700
// MI455X (gfx1250) — compile-verified
//
#include <hip/hip_runtime.h>
#include <hip/hip_bf16.h>

// upfirdn2d(up=2, down=1, pad=(2,1), k=[1,3,3,1] outer) on (16,512,64,64) f32.
// Separable polyphase form (kernel is palindromic, so the reference's flip is
// identity):
//   out[2i]   = 0.25*x[i-1] + 0.75*x[i]
//   out[2i+1] = 0.75*x[i]   + 0.25*x[i+1]
// per dimension, with zero outside [0,64).
//
// Memory-bound (≈1.5 flop/byte): 128MB in + 512MB out -> ~28us floor at
// 23.3 TB/s. Strategy: async global->LDS staging (CDNA5 TDM-class path,
// ASYNCcnt), compute from LDS, stream results out with non-temporal b64
// stores so the 512MB output stream does not thrash the 192MB L2 (the whole
// input fits in L2).

#define IN_H 64
#define IN_W 64
#define TILE_IN_ROWS 18   // 16 interior rows + 1 halo row above/below
#define TILE_IN_COLS 66   // 64 interior cols + 1 halo col left/right
#define LDS_STRIDE   68   // padded row stride (floats)

typedef float v2f __attribute__((ext_vector_type(2)));

__global__ __launch_bounds__(256) void UpsampleF_89584427860700_kernel(
    const float* __restrict__ src,   // (N*C, 64, 64)
    float* __restrict__ dst)         // (N*C, 128, 128)
{
    __shared__ float tile[TILE_IN_ROWS * LDS_STRIDE];

    const int img = blockIdx.y;      // image index in [0, N*C)
    const int i0  = blockIdx.x * 16; // first interior input row of this tile
    const int tid = threadIdx.x;     // 0..255

    const float* srcImg = src + (size_t)img * (IN_H * IN_W);

    // ---- Phase 1: async-copy input tile (+zero halo) into LDS -------------
    // tile row r  <-> input row  i0 - 1 + r
    // tile col c  <-> input col  c - 1
    for (int s = tid; s < TILE_IN_ROWS * TILE_IN_COLS; s += 256) {
        const int r  = s / TILE_IN_COLS;
        const int c  = s - r * TILE_IN_COLS;
        const int gr = i0 - 1 + r;
        const int gc = c - 1;
        // Low 32 bits of a generic pointer into LDS == LDS byte address.
        const unsigned lds_addr =
            (unsigned)(uintptr_t)(&tile[r * LDS_STRIDE + c]);
        if ((unsigned)gr < (unsigned)IN_H && (unsigned)gc < (unsigned)IN_W) {
            const unsigned long long ga =
                (unsigned long long)(uintptr_t)(srcImg + gr * IN_W + gc);
            // VDST = LDS byte address, VADDR = 64-bit global address (GV mode).
            asm volatile("global_load_async_to_lds_b32 %0, %1, off"
                         :: "v"(lds_addr), "v"(ga)
                         : "memory");
        } else {
            tile[r * LDS_STRIDE + c] = 0.0f;   // zero pad region
        }
    }
    // Async loads are tracked by ASYNCcnt, invisible to the compiler's
    // ds/load counters -> drain explicitly before the workgroup barrier.
    asm volatile("s_wait_asynccnt 0" ::: "memory");
    __syncthreads();

    // ---- Phase 2: each thread computes four 2x2 output blocks -------------
    const int jj  = tid & 63;   // interior input col 0..63 -> out cols 2jj,2jj+1
    const int ii0 = tid >> 6;   // 0..3
    float* dstImg = dst + (size_t)img * (2 * IN_H) * (2 * IN_W);

    #pragma unroll
    for (int k = 0; k < 4; ++k) {
        const int ii = ii0 + 4 * k;               // interior input row 0..15
        // tile rows: ii -> input row i-1, ii+1 -> i, ii+2 -> i+1
        const float* t0 = &tile[(ii + 0) * LDS_STRIDE + jj];
        const float* t1 = &tile[(ii + 1) * LDS_STRIDE + jj];
        const float* t2 = &tile[(ii + 2) * LDS_STRIDE + jj];

        const float a0 = t0[0], a1 = t0[1], a2 = t0[2];   // row i-1
        const float b0 = t1[0], b1 = t1[1], b2 = t1[2];   // row i
        const float c0 = t2[0], c1 = t2[1], c2 = t2[2];   // row i+1

        // Horizontal polyphase (even/odd output column) for the 3 rows.
        const float he_m = 0.25f * a0 + 0.75f * a1;
        const float ho_m = 0.75f * a1 + 0.25f * a2;
        const float he_0 = 0.25f * b0 + 0.75f * b1;
        const float ho_0 = 0.75f * b1 + 0.25f * b2;
        const float he_p = 0.25f * c0 + 0.75f * c1;
        const float ho_p = 0.75f * c1 + 0.25f * c2;

        // Vertical polyphase -> 2x2 output block.
        v2f row0, row1;
        row0.x = 0.25f * he_m + 0.75f * he_0;   // out(2i,   2j)
        row0.y = 0.25f * ho_m + 0.75f * ho_0;   // out(2i,   2j+1)
        row1.x = 0.75f * he_0 + 0.25f * he_p;   // out(2i+1, 2j)
        row1.y = 0.75f * ho_0 + 0.25f * ho_p;   // out(2i+1, 2j+1)

        const int orow = 2 * (i0 + ii);
        v2f* p0 = (v2f*)(dstImg + (size_t)orow       * (2 * IN_W) + 2 * jj);
        v2f* p1 = (v2f*)(dstImg + (size_t)(orow + 1) * (2 * IN_W) + 2 * jj);
        // Streaming output: non-temporal so the 512MB write stream does not
        // evict the (L2-resident, 128MB) input.
        __builtin_nontemporal_store(row0, p0);
        __builtin_nontemporal_store(row1, p1);
    }
}

extern "C" void kernel_launch(void* const* d_in, const int* in_sizes, int n_in,
                              void* d_out, int out_size, void* d_ws, size_t ws_size,
                              hipStream_t stream) {
    const float* fmap = (const float*)d_in[0];
    // d_in[1] is the 4x4 FIR kernel; it is the fixed normalized [1,3,3,1]
    // outer product, folded into the 0.25/0.75 polyphase constants above.
    float* out = (float*)d_out;

    const int NC = in_sizes[0] / (IN_H * IN_W);   // 16*512 = 8192 images
    dim3 grid(IN_H / 16, NC);                     // 4 row-tiles x 8192 images
    UpsampleF_89584427860700_kernel<<<grid, 256, 0, stream>>>(fmap, out);
}